// GLT_43516608643674
// MI455X (gfx1250) — compile-verified
//
#include <hip/hip_runtime.h>

#define Bv 8
#define Lv 64
#define Hv 128

typedef float v2f __attribute__((ext_vector_type(2)));
typedef float v8f __attribute__((ext_vector_type(8)));

// ---------------------------------------------------------------------------
// Chart init: T[:,0] = words_feats, everything else 0. Re-run every call.
// ---------------------------------------------------------------------------
__global__ void init_chart(float* __restrict__ T, const float* __restrict__ feats) {
    int idx = blockIdx.x * blockDim.x + threadIdx.x;
    const int total = Bv * Lv * Lv * Hv;
    if (idx >= total) return;
    int h = idx % Hv;
    int j = (idx / Hv) % Lv;
    int r = (idx / (Hv * Lv)) % Lv;
    int b = idx / (Hv * Lv * Lv);
    float v = 0.0f;
    if (r == 0) v = feats[(b * Lv + j) * Hv + h];
    T[idx] = v;
}

// ---------------------------------------------------------------------------
// One recursion step i. Grid: (ceil((L-i)/16), B), block: 128 threads (4 waves).
// Each block: batch b, 16 consecutive j columns; loops k = 0..i-1 with online
// softmax over k. 128x128 dense layer done with V_WMMA_F32_16X16X4_F32.
// ---------------------------------------------------------------------------
__launch_bounds__(128)
__global__ void step_kernel(float* __restrict__ T, int i,
                            const float* __restrict__ Wl, const float* __restrict__ bl,
                            const float* __restrict__ Wr, const float* __restrict__ br,
                            const float* __restrict__ Wrep, const float* __restrict__ brep,
                            const float* __restrict__ Ws, const float* __restrict__ bs,
                            float* __restrict__ out) {
    // +4 pad on 16-row tiles -> conflict-free ds_load_b64 A-fragment reads
    __shared__ float s_hhat[16 * 132];
    __shared__ float s_temp[16 * 132];
    __shared__ float s_acc[16 * 128];
    __shared__ float s_redl[128];
    __shared__ float s_redr[128];
    __shared__ float s_w0[16], s_w1[16];
    __shared__ float s_m[16], s_s[16], s_scale[16], s_p[16];

    const int tid  = threadIdx.x;
    const int lane = tid & 31;
    const int wv   = tid >> 5;          // wave id 0..3 -> N columns [32*wv, 32*wv+32)
    const int b    = blockIdx.y;
    const int j0   = blockIdx.x * 16;
    const int njv  = Lv - i;            // valid j: 0..njv-1

    const int jl = tid >> 3;            // 0..15 : j row handled in reductions
    const int hb = (tid & 7) << 4;      // 16-wide h chunk

    // --- preload Wrep B-fragments (f32 4x16 per K-step) into registers -----
    // B layout guess per ISA: VGPR0 = B[K=0|2][n], VGPR1 = B[K=1|3][n],
    // lanes 0-15: K-lo half (n = lane), lanes 16-31: K-hi half (same n).
    const int khi = lane >> 4;
    const int nb  = (lane & 15) + wv * 32;
    v2f frag[2][32];
#pragma unroll
    for (int ks = 0; ks < 32; ++ks) {
        const int kr = ks * 4 + 2 * khi;
#pragma unroll
        for (int nt = 0; nt < 2; ++nt) {
            const int n = nb + nt * 16;
            v2f f;
            f.x = Wrep[kr * Hv + n];
            f.y = Wrep[(kr + 1) * Hv + n];
            frag[nt][ks] = f;
        }
    }
    const float brep0 = brep[nb];
    const float brep1 = brep[nb + 16];
    const float blv = bl[0], brv = br[0], bsv = bs[0];

    // --- online-softmax init: invalid k (k>=i) each contribute exp(bs) -----
    if (tid < 16) { s_m[tid] = bsv; s_s[tid] = (float)(Lv - i); }
    for (int q = 0; q < 16; ++q) s_acc[q * 128 + tid] = 0.0f;
    __syncthreads();

    for (int k = 0; k < i; ++k) {
        // ---------------- phase 1: gates + h_hat -> LDS --------------------
        const int  jglob  = j0 + jl;
        const bool jvalid = jglob < njv;
        const int  jc     = jvalid ? jglob : 0;
        const float* pl = T + ((size_t)(b * Lv + k) * Lv + jc) * Hv + hb;
        const float* pr = T + ((size_t)(b * Lv + (i - k - 1)) * Lv + (jc + k + 1)) * Hv + hb;

        float lv[16], rv[16];
        float dl = 0.0f, dr = 0.0f;
#pragma unroll
        for (int e = 0; e < 16; ++e) {
            float a_ = jvalid ? pl[e] : 0.0f;
            float c_ = jvalid ? pr[e] : 0.0f;
            lv[e] = a_; rv[e] = c_;
            dl += a_ * Wl[hb + e];
            dr += c_ * Wr[hb + e];
        }
        s_redl[jl * 8 + (tid & 7)] = dl;
        s_redr[jl * 8 + (tid & 7)] = dr;
        __syncthreads();
        if (tid < 16) {
            float sl = blv, sr = brv;
            for (int q = 0; q < 8; ++q) { sl += s_redl[tid * 8 + q]; sr += s_redr[tid * 8 + q]; }
            float mx  = fmaxf(sl, sr);
            float e0  = __expf(sl - mx), e1 = __expf(sr - mx);
            float inv = 1.0f / (e0 + e1);
            s_w0[tid] = e0 * inv;
            s_w1[tid] = e1 * inv;
        }
        __syncthreads();
        {
            const float w0 = s_w0[jl], w1 = s_w1[jl];
#pragma unroll
            for (int e = 0; e < 16; ++e)
                s_hhat[jl * 132 + hb + e] = w0 * lv[e] + w1 * rv[e];
        }
        __syncthreads();

        // ---- phase 2: temp = relu(h_hat @ Wrep + brep) + h_hat  (WMMA) ----
        v8f c0 = {0, 0, 0, 0, 0, 0, 0, 0};
        v8f c1 = {0, 0, 0, 0, 0, 0, 0, 0};
        const int arow = lane & 15;
#pragma unroll
        for (int ks = 0; ks < 32; ++ks) {
            // A fragment: lane m -> row m, cols 4*ks + 2*khi + {0,1}
            v2f a = *(const v2f*)&s_hhat[arow * 132 + ks * 4 + 2 * khi];
            c0 = __builtin_amdgcn_wmma_f32_16x16x4_f32(false, a, false, frag[0][ks],
                                                       (short)0, c0, false, false);
            c1 = __builtin_amdgcn_wmma_f32_16x16x4_f32(false, a, false, frag[1][ks],
                                                       (short)0, c1, false, false);
        }
        // C/D layout: VGPR p -> rows p (lanes 0-15) / p+8 (lanes 16-31), N = lane&15
#pragma unroll
        for (int p = 0; p < 8; ++p) {
            const int row = p + (khi ? 8 : 0);
            float v0 = c0[p] + brep0;
            v0 = fmaxf(v0, 0.0f) + s_hhat[row * 132 + nb];
            s_temp[row * 132 + nb] = v0;
            float v1 = c1[p] + brep1;
            v1 = fmaxf(v1, 0.0f) + s_hhat[row * 132 + nb + 16];
            s_temp[row * 132 + nb + 16] = v1;
        }
        __syncthreads();

        // ---------------- phase 3: logits + online softmax -----------------
        float pq = 0.0f;
#pragma unroll
        for (int e = 0; e < 16; ++e) pq += s_temp[jl * 132 + hb + e] * Ws[hb + e];
        s_redl[jl * 8 + (tid & 7)] = pq;
        __syncthreads();
        if (tid < 16) {
            float logit = bsv;
            for (int q = 0; q < 8; ++q) logit += s_redl[tid * 8 + q];
            float m  = s_m[tid];
            float mn = fmaxf(m, logit);
            float sc = __expf(m - mn);
            float pp = __expf(logit - mn);
            s_s[tid]     = s_s[tid] * sc + pp;
            s_m[tid]     = mn;
            s_scale[tid] = sc;
            s_p[tid]     = pp;
        }
        __syncthreads();

        // ---------------- phase 4: accumulator update ----------------------
        for (int q = 0; q < 16; ++q)
            s_acc[q * 128 + tid] = s_acc[q * 128 + tid] * s_scale[q]
                                 + s_p[q] * s_temp[q * 132 + tid];
        __syncthreads();
    }

    // ---------------- epilogue: write row i (and output at i = L-1) --------
    for (int q = 0; q < 16; ++q) {
        const int j = j0 + q;
        if (j < njv) {
            const float v = s_acc[q * 128 + tid] / s_s[q];
            T[((size_t)(b * Lv + i) * Lv + j) * Hv + tid] = v;
            if (i == Lv - 1 && j == 0) out[b * Hv + tid] = v;
        }
    }
}

// ---------------------------------------------------------------------------
extern "C" void kernel_launch(void* const* d_in, const int* in_sizes, int n_in,
                              void* d_out, int out_size, void* d_ws, size_t ws_size,
                              hipStream_t stream) {
    const float* feats = (const float*)d_in[0];
    // d_in[1] = words_masks (all ones, unused by reference math)
    const float* Wl   = (const float*)d_in[2];
    const float* bl   = (const float*)d_in[3];
    const float* Wr   = (const float*)d_in[4];
    const float* br   = (const float*)d_in[5];
    const float* Wrep = (const float*)d_in[6];
    const float* brep = (const float*)d_in[7];
    const float* Ws   = (const float*)d_in[8];
    const float* bs   = (const float*)d_in[9];

    float* T   = (float*)d_ws;                 // B*L*L*H f32 chart = 16 MB
    float* out = (float*)d_out;

    {
        const int total = Bv * Lv * Lv * Hv;
        init_chart<<<(total + 255) / 256, 256, 0, stream>>>(T, feats);
    }
    for (int i = 1; i < Lv; ++i) {
        const int njv = Lv - i;
        dim3 grid((njv + 15) / 16, Bv);
        step_kernel<<<grid, 128, 0, stream>>>(T, i, Wl, bl, Wr, br, Wrep, brep,
                                              Ws, bs, out);
    }
}